// ScaledDotProductAttention_11441792877149
// MI455X (gfx1250) — compile-verified
//
#include <hip/hip_runtime.h>

typedef __attribute__((ext_vector_type(2))) float v2f;
typedef __attribute__((ext_vector_type(8))) float v8f;

#define BS   4
#define SEQ  4096
#define DK   128
#define KTILES (SEQ / 16)          // 256
#define SCALE 0.08838834764831845f // 1/sqrt(128)
#define WAVES_PER_BLOCK 8

// D = A(16x4 f32) * B(4x16 f32) + C(16x16 f32)
__device__ __forceinline__ v8f wmma_f32(v2f a, v2f b, v8f c) {
  return __builtin_amdgcn_wmma_f32_16x16x4_f32(
      /*neg_a=*/false, a, /*neg_b=*/false, b,
      /*c_mod=*/(short)0, c, /*reuse_a=*/false, /*reuse_b=*/false);
}

// -------------------------------------------------------------------------
// Pass 1: per-(b,k) column max and sum-of-exp of the scaled score matrix,
// streaming q-tiles with an online-softmax update. One wave per 16 columns.
// -------------------------------------------------------------------------
__global__ __launch_bounds__(256)
void sdpa_colstats_kernel(const float* __restrict__ Q,
                          const float* __restrict__ K,
                          float* __restrict__ mArr,
                          float* __restrict__ lArr) {
  const int lane  = threadIdx.x & 31;
  const int wid   = blockIdx.x * WAVES_PER_BLOCK + (threadIdx.x >> 5);
  const int b     = wid >> 8;            // wid / KTILES
  const int kt    = (wid & 255) << 4;    // key-tile base
  const int n     = lane & 15;           // column within tile
  const int khalf = (lane >> 4) << 1;    // K-depth sub-offset {0,2}

  // B operand: 16 key rows, full DK depth, held in registers (32 x float2).
  const float* Krow = K + ((size_t)b * SEQ + kt + n) * DK + khalf;
  v2f breg[32];
#pragma unroll
  for (int c = 0; c < 32; ++c) breg[c] = *(const v2f*)(Krow + 4 * c);

  const float* Qb = Q + (size_t)b * SEQ * DK;
  float m_run = -__builtin_inff();
  float l_run = 0.0f;

  for (int qt = 0; qt < SEQ; qt += 16) {
    const float* Qrow = Qb + (size_t)(qt + n) * DK + khalf;
    v8f acc = {};
#pragma unroll
    for (int c = 0; c < 32; ++c) {
      v2f a = *(const v2f*)(Qrow + 4 * c);
      acc = wmma_f32(a, breg[c], acc);
    }
    // C layout: this lane holds column n, rows M = j + 8*(lane>=16).
    float mloc = acc[0] * SCALE;
#pragma unroll
    for (int j = 1; j < 8; ++j) mloc = fmaxf(mloc, acc[j] * SCALE);
    const float m_new = fmaxf(m_run, mloc);
    float s = 0.0f;
#pragma unroll
    for (int j = 0; j < 8; ++j) s += __expf(acc[j] * SCALE - m_new);
    l_run = l_run * __expf(m_run - m_new) + s;
    m_run = m_new;
  }

  // Merge the two lane-halves (rows 0..7 vs 8..15 of every q-tile).
  const float m_o = __shfl_xor(m_run, 16, 32);
  const float l_o = __shfl_xor(l_run, 16, 32);
  const float m_f = fmaxf(m_run, m_o);
  const float l_f = l_run * __expf(m_run - m_f) + l_o * __expf(m_o - m_f);
  if (lane < 16) {
    mArr[(size_t)b * SEQ + kt + n] = m_f;
    lArr[(size_t)b * SEQ + kt + n] = l_f;
  }
}

// -------------------------------------------------------------------------
// Pass 2: O[qtile,:] = sum_k exp(S - m[k]) / l[k] * V[k,:]
// One wave per 16 query rows; recompute S tiles with WMMA, transpose P
// through a wave-private LDS tile, second WMMA chain into 16x128 acc.
// -------------------------------------------------------------------------
__global__ __launch_bounds__(256)
void sdpa_out_kernel(const float* __restrict__ Q,
                     const float* __restrict__ K,
                     const float* __restrict__ V,
                     const float* __restrict__ mArr,
                     const float* __restrict__ lArr,
                     float* __restrict__ O) {
  __shared__ float smem[WAVES_PER_BLOCK][16 * 17];

  const int lane  = threadIdx.x & 31;
  const int wslot = threadIdx.x >> 5;
  const int wid   = blockIdx.x * WAVES_PER_BLOCK + wslot;
  const int b     = wid >> 8;
  const int qt    = (wid & 255) << 4;
  const int n     = lane & 15;
  const int khalf = (lane >> 4) << 1;
  const int mhi   = (lane >> 4) << 3;   // +8 row offset for upper lane-half

  // Q tile in A-format, resident for the whole kernel (32 x float2).
  const float* Qrow = Q + ((size_t)b * SEQ + qt + n) * DK + khalf;
  v2f areg[32];
#pragma unroll
  for (int c = 0; c < 32; ++c) areg[c] = *(const v2f*)(Qrow + 4 * c);

  const v8f vzero = {};
  v8f acc[8];                            // 16 q-rows x 128 d (8 d-tiles)
#pragma unroll
  for (int t = 0; t < 8; ++t) acc[t] = vzero;

  const float* Kb = K + (size_t)b * SEQ * DK;
  const float* Vb = V + (size_t)b * SEQ * DK;
  const float* mB = mArr + (size_t)b * SEQ;
  const float* lB = lArr + (size_t)b * SEQ;
  float* sm = &smem[wslot][0];

  for (int kt = 0; kt < SEQ; kt += 16) {
    // --- S tile = Q_tile * K_tile^T (32 chained f32 WMMAs) ---
    const float* Krow = Kb + (size_t)(kt + n) * DK + khalf;
    v8f c8 = vzero;
#pragma unroll
    for (int c = 0; c < 32; ++c) {
      v2f bk = *(const v2f*)(Krow + 4 * c);
      c8 = wmma_f32(areg[c], bk, c8);
    }

    // --- P = exp(S*scale - m[col]) / l[col], column-normalized ---
    const float mcol = mB[kt + n];
    const float rl   = 1.0f / lB[kt + n];
#pragma unroll
    for (int j = 0; j < 8; ++j) {
      const float p = __expf(c8[j] * SCALE - mcol) * rl;
      sm[(j + mhi) * 17 + n] = p;        // store [q_row][k_col]
    }
    __builtin_amdgcn_wave_barrier();     // keep ds_store before ds_load

    // --- re-read P in A-format: lane row M=n, K-depth 4*c2+khalf ---
    v2f pa[4];
#pragma unroll
    for (int c2 = 0; c2 < 4; ++c2) {
      const int kk = 4 * c2 + khalf;
      pa[c2].x = sm[n * 17 + kk];
      pa[c2].y = sm[n * 17 + kk + 1];
    }

    // --- O += P * V (8 d-tiles x 4 K-chunks = 32 WMMAs) ---
#pragma unroll
    for (int t = 0; t < 8; ++t) {
      const float* Vcol = Vb + (size_t)(kt + khalf) * DK + t * 16 + n;
#pragma unroll
      for (int c2 = 0; c2 < 4; ++c2) {
        v2f bv;
        bv.x = Vcol[(size_t)(4 * c2) * DK];
        bv.y = Vcol[(size_t)(4 * c2 + 1) * DK];
        acc[t] = wmma_f32(pa[c2], bv, acc[t]);
      }
    }
  }

  // --- write O: element (q = qt + j + mhi, d = 16t + n) ---
  float* Ob = O + ((size_t)b * SEQ + qt) * DK;
#pragma unroll
  for (int t = 0; t < 8; ++t) {
#pragma unroll
    for (int j = 0; j < 8; ++j) {
      Ob[(size_t)(j + mhi) * DK + t * 16 + n] = acc[t][j];
    }
  }
}

// -------------------------------------------------------------------------
extern "C" void kernel_launch(void* const* d_in, const int* in_sizes, int n_in,
                              void* d_out, int out_size, void* d_ws, size_t ws_size,
                              hipStream_t stream) {
  const float* q = (const float*)d_in[0];
  const float* k = (const float*)d_in[1];
  const float* v = (const float*)d_in[2];
  float* out = (float*)d_out;

  float* mArr = (float*)d_ws;                 // BS*SEQ floats
  float* lArr = mArr + (size_t)BS * SEQ;      // BS*SEQ floats

  const int total_waves = BS * KTILES;                      // 1024
  const int blocks = total_waves / WAVES_PER_BLOCK;         // 128
  const int threads = WAVES_PER_BLOCK * 32;                 // 256

  sdpa_colstats_kernel<<<blocks, threads, 0, stream>>>(q, k, mArr, lArr);
  sdpa_out_kernel<<<blocks, threads, 0, stream>>>(q, k, v, mArr, lArr, out);
}